// SE3TransformerBlock_84894323573079
// MI455X (gfx1250) — compile-verified
//
#include <hip/hip_runtime.h>
#include <math.h>

// ---------------------------------------------------------------------------
// SE(3)-transformer block for MI455X (gfx1250, wave32, WMMA).
// Sizes fixed by setup_inputs(): B=4, N=192, CS=16, CV=4, DIM=28, NUMEL=400.
//
// G matrices and the LDS h buffer are stored K-pair interleaved so every WMMA
// operand (K-rows t0,t0+1 for one lane) is a single aligned 8-byte load.
//   G index: (b*N+j)*1280 + (t/2)*160 + col*2 + (t&1)
//   h index: mlp*4096 + jj*256 + (t/2)*32 + i*2 + (t&1)
// ---------------------------------------------------------------------------

typedef float v2f __attribute__((ext_vector_type(2)));
typedef float v8f __attribute__((ext_vector_type(8)));

#define BB      4
#define NN      192
#define DIMF    28
#define JCHUNKS 4          // j split into 4 chunks of 48 (3 tiles of 16)
#define JT_PER  3
#define GSTRIDE 1280       // 16 t-rows * 80 cols per (b,j) per MLP
#define YP      81         // padded pitch for y tile in LDS

// scale constants
#define S_A   0.17677669529663687f  /* 1/(4*sqrt(2))   */
#define S_B   0.2041241452319315f   /* 1/sqrt(24)      */
#define S_V2  0.35355339059327373f  /* 1/(2*sqrt(2))   */

__device__ __forceinline__ float sus_f(float x) {
  return (x > 0.0f) ? expf(-1.0f / x) : 0.0f;
}

// ---------------------------------------------------------------------------
// Kernel 1: fold neighbor features fs/fv into W2 -> per-j matrices G (16x80).
//   cols  0..15 : sum_c fs[c]  * W2[t, c*16+o]        (w1 block -> scalar out)
//   cols 16..63 : sum_c fv[c,x]* W2[t, 336+c*16+o]    (w4 block, x=0..2)
//   cols 64..67 : sum_c fs[c]  * W2[t, 256+c*4+o]     (w2 block -> vector out)
//   cols 68..79 : sum_c fv[c,x]* W2[t, 320+c*4+o]     (w3 block, x-major by 4)
// Stored K-pair interleaved (see header comment).
// ---------------------------------------------------------------------------
__global__ void k_prep_g(const float* __restrict__ f,
                         const float* __restrict__ w2k,
                         const float* __restrict__ w2v,
                         float* __restrict__ Gk, float* __restrict__ Gv) {
  int idx = blockIdx.x * blockDim.x + threadIdx.x;
  if (idx >= BB * NN * 16 * 80) return;
  int col = idx % 80; int tmp = idx / 80;
  int t = tmp % 16;  tmp /= 16;
  int j = tmp % NN;  int b = tmp / NN;
  const float* fr = f + (size_t)(b * NN + j) * DIMF;
  const float* rk = w2k + t * 400;
  const float* rv = w2v + t * 400;
  float ak = 0.f, av = 0.f;
  if (col < 16) {
    int o = col;
    for (int c = 0; c < 16; ++c) { float s = fr[c]; int m = c*16+o; ak += s*rk[m]; av += s*rv[m]; }
  } else if (col < 64) {
    int x = (col - 16) >> 4, o = (col - 16) & 15;
    for (int c = 0; c < 4; ++c) { float s = fr[16 + c*3 + x]; int m = 336 + c*16 + o; ak += s*rk[m]; av += s*rv[m]; }
  } else if (col < 68) {
    int o = col - 64;
    for (int c = 0; c < 16; ++c) { float s = fr[c]; int m = 256 + c*4 + o; ak += s*rk[m]; av += s*rv[m]; }
  } else {
    int x = (col - 68) >> 2, o = (col - 68) & 3;
    for (int c = 0; c < 4; ++c) { float s = fr[16 + c*3 + x]; int m = 320 + c*4 + o; ak += s*rk[m]; av += s*rv[m]; }
  }
  size_t st = (size_t)(b * NN + j) * GSTRIDE + (t >> 1) * 160 + col * 2 + (t & 1);
  Gk[st] = ak; Gv[st] = av;
}

// ---------------------------------------------------------------------------
// Kernel 2: per-query precompute: qds' = (q_s . dot_ws)/(4*sqrt2),
//           qdv' = (q_v . dot_wv)/sqrt(24), plus validity flags.
// ---------------------------------------------------------------------------
__global__ void k_prep_q(const float* __restrict__ f,
                         const float* __restrict__ wq_s,
                         const float* __restrict__ wq_v,
                         const float* __restrict__ dot_ws,
                         const float* __restrict__ dot_wv,
                         float* __restrict__ qds, float* __restrict__ qdv,
                         float* __restrict__ validv) {
  int t = blockIdx.x * 256 + threadIdx.x;
  if (t >= BB * NN) return;
  const float* fr = f + (size_t)t * DIMF;
  float v = 0.f;
  for (int c = 0; c < DIMF; ++c) if (fr[c] != 0.f) v = 1.f;
  validv[t] = v;
  float qs[16];
  for (int c = 0; c < 16; ++c) {
    float a = 0.f;
    for (int e = 0; e < 16; ++e) a += fr[e] * wq_s[e*16 + c];
    qs[c] = 0.25f * a;                              // /sqrt(CS)
  }
  for (int d = 0; d < 16; ++d) {
    float a = 0.f;
    for (int c = 0; c < 16; ++c) a += qs[c] * dot_ws[c*16 + d];
    qds[(size_t)t*16 + d] = S_A * a;                // /(sqrt(CS)*sqrt(2))
  }
  float qv[4][3];
  for (int x = 0; x < 3; ++x)
    for (int c = 0; c < 4; ++c) {
      float a = 0.f;
      for (int e = 0; e < 4; ++e) a += fr[16 + e*3 + x] * wq_v[e*4 + c];
      qv[c][x] = 0.5f * a;                          // /sqrt(CV)
    }
  for (int d = 0; d < 4; ++d)
    for (int x = 0; x < 3; ++x) {
      float a = 0.f;
      for (int c = 0; c < 4; ++c) a += qv[c][x] * dot_wv[c*4 + d];
      qdv[(size_t)t*12 + d*3 + x] = S_B * a;        // /(sqrt(3CV)*sqrt(2))
    }
}

// ---------------------------------------------------------------------------
// WMMA pass: for each j in tile, y[jj] = h[jj](16i x 16t) @ G[j](16t x 80col)
// using chained V_WMMA_F32_16X16X4_F32 (K = 4*4 = 16).
// A layout (16x4 f32): lanes 0-15 -> M=lane, VGPR0=K0,VGPR1=K1;
//                      lanes 16-31 -> M=lane-16, VGPR0=K2,VGPR1=K3.
// B layout (4x16 f32): lanes 0-15 -> N=lane, VGPR0=K0,VGPR1=K1;
//                      lanes 16-31 -> N=lane-16, VGPR0=K2,VGPR1=K3.
// D layout (16x16 f32): VGPR r -> M=r (lanes 0-15) / M=8+r (lanes 16-31), N=lane%16.
// Both h (LDS) and G (global) are K-pair packed -> one b64 load per operand.
// ---------------------------------------------------------------------------
__device__ __forceinline__ void wmma_pass(const float* __restrict__ Gj0,
                                          const float* __restrict__ sh,
                                          float* __restrict__ sy,
                                          int lane, int wave) {
  const int n = lane & 15, half = lane >> 4;
  for (int tk = wave; tk < 80; tk += 8) {      // 16 j-slots * 5 col-tiles
    const int jj = tk / 5, ct = tk % 5;
    const float* Gj = Gj0 + (size_t)jj * GSTRIDE;
    v8f c = {};
#pragma unroll
    for (int kk = 0; kk < 4; ++kk) {
      const int kp = kk * 2 + half;            // K-pair index, t0 = 2*kp
      v2f a  = *(const v2f*)&sh[jj*256 + kp*32 + n*2];
      v2f bm = *(const v2f*)&Gj[kp*160 + (ct*16 + n)*2];
      c = __builtin_amdgcn_wmma_f32_16x16x4_f32(false, a, false, bm,
                                                (short)0, c, false, false);
    }
#pragma unroll
    for (int r = 0; r < 8; ++r) {
      const int i = r + 8 * half;
      sy[(jj*16 + i)*YP + ct*16 + n] = c[r];
    }
  }
}

// ---------------------------------------------------------------------------
// Kernel 3: main attention. Grid = (jc=4, it=12, b=4), 256 threads (8 waves).
// One 16-query tile, loops over 3 j-tiles of its 48-j chunk, single pass:
//   z += log_exp; accP += sqrt(|log_exp|)*v (log_exp>0);  accN likewise (<0).
// ---------------------------------------------------------------------------
__global__ __launch_bounds__(256) void k_attn(
    const float* __restrict__ pos,
    const float* __restrict__ w1k, const float* __restrict__ w1v,
    const float* __restrict__ Gk,  const float* __restrict__ Gv,
    const float* __restrict__ qds_g, const float* __restrict__ qdv_g,
    const float* __restrict__ valid_g,
    float* __restrict__ zpart, float* __restrict__ outP, float* __restrict__ outN) {
  __shared__ float s_h[2 * 16 * 16 * 16];   // [mlp][jj][t/2][i][t&1]
  __shared__ float s_y[16 * 16 * YP];       // [jj][i][col(80, pitch 81)]
  __shared__ float s_geo[256 * 4];          // sh1.xyz, cutoff per pair
  __shared__ float s_le[256];               // log_exp per pair
  __shared__ float s_accP[16 * 28], s_accN[16 * 28], s_accZ[16];
  __shared__ float s_qds[256], s_qdv[192], s_w1[320];

  const int p  = threadIdx.x;
  const int jc = blockIdx.x, it = blockIdx.y, b = blockIdx.z;
  const int i0 = it * 16;
  const int lane = p & 31, wave = p >> 5;
  const int bN = b * NN;

  for (int q = p; q < 16 * 28; q += 256) { s_accP[q] = 0.f; s_accN[q] = 0.f; }
  if (p < 16) s_accZ[p] = 0.f;
  { int i = p >> 4, d = p & 15; s_qds[p] = qds_g[(size_t)(bN + i0 + i)*16 + d]; }
  if (p < 192) { int i = p / 12, r = p % 12; s_qdv[p] = qdv_g[(size_t)(bN + i0 + i)*12 + r]; }
  if (p < 320) { int fc = p / 160, r = p % 160; s_w1[p] = (fc ? w1v : w1k)[r]; }
  __syncthreads();

  const int ii = p & 15, jj = p >> 4;
  const int gi = i0 + ii;
  const float vi  = valid_g[bN + gi];
  const float pix = pos[(bN + gi)*3 + 0];
  const float piy = pos[(bN + gi)*3 + 1];
  const float piz = pos[(bN + gi)*3 + 2];
  const float EMB_GAIN = 1.14136f * 7.3890560989f * 3.1622776601f;

  for (int jt = 0; jt < JT_PER; ++jt) {
    const int j0 = jc * 48 + jt * 16;
    const int gj = j0 + jj;
    // ---- prefetch next tile's G stream into caches (global_prefetch_b8) ----
    if (jt + 1 < JT_PER) {
      const float* nk = Gk + (size_t)(bN + j0 + 16) * GSTRIDE;
      const float* nv = Gv + (size_t)(bN + j0 + 16) * GSTRIDE;
      __builtin_prefetch(nk + (size_t)p * 80, 0, 1);
      __builtin_prefetch(nv + (size_t)p * 80, 0, 1);
    }
    // ---- phase 0: geometry, radial basis, both MLP hidden layers -----------
    {
      float dx = pos[(bN + gj)*3 + 0] - pix;
      float dy = pos[(bN + gj)*3 + 1] - piy;
      float dz = pos[(bN + gj)*3 + 2] - piz;
      float dist = sqrtf(dx*dx + dy*dy + dz*dz + 1e-12f);
      float inv = 1.7320508075688772f / dist;       // sqrt(3)/dist
      float cut = (dist < 1.5f && gi != gj && vi != 0.f && valid_g[bN + gj] != 0.f)
                      ? sus_f(10.f * (1.f - dist * (1.f / 1.5f))) : 0.f;
      s_geo[p*4 + 0] = dx * inv; s_geo[p*4 + 1] = dy * inv;
      s_geo[p*4 + 2] = dz * inv; s_geo[p*4 + 3] = cut;
      float emb[10];
      float dd = dist * 7.3333333333333333f;        // dist/step, step=1.5/11
      for (int k = 0; k < 10; ++k)
        emb[k] = EMB_GAIN * sus_f(dd - (float)k) * sus_f((float)(k + 2) - dd);
      for (int fc = 0; fc < 2; ++fc) {
        const float* W1 = &s_w1[fc * 160];
        for (int t = 0; t < 16; ++t) {
          float u = 0.f;
          for (int k = 0; k < 10; ++k) u += emb[k] * W1[k*16 + t];
          u *= 0.31622776601683794f;                // /sqrt(NUM_BASIS)
          float h = 0.419750f * u / (1.f + expf(-u)); // SILU_GAIN*silu * 0.25
          s_h[fc*4096 + jj*256 + (t >> 1)*32 + ii*2 + (t & 1)] = h;
        }
      }
    }
    __syncthreads();
    // ---- phase A: keys (WMMA), score, z ------------------------------------
    wmma_pass(Gk + (size_t)(bN + j0) * GSTRIDE, &s_h[0], s_y, lane, wave);
    __syncthreads();
    {
      const float* y = &s_y[(jj*16 + ii) * YP];
      float sh0 = s_geo[p*4], sh1 = s_geo[p*4+1], sh2 = s_geo[p*4+2], cut = s_geo[p*4+3];
      float shv[3] = {sh0, sh1, sh2};
      float sc = 0.f;
      for (int o = 0; o < 16; ++o) {
        float ks = S_A * y[o] + S_B * (sh0*y[16+o] + sh1*y[32+o] + sh2*y[48+o]);
        sc += s_qds[ii*16 + o] * ks;
      }
      for (int o = 0; o < 4; ++o) {
        float ya = y[64 + o];
        for (int x = 0; x < 3; ++x) {
          float kv = S_A * shv[x] * ya + S_V2 * y[68 + 4*x + o];
          sc += s_qdv[ii*12 + o*3 + x] * kv;
        }
      }
      float le = cut * sc;
      s_le[p] = le;
      atomicAdd(&s_accZ[ii], le);
    }
    __syncthreads();
    // ---- phase B: values (WMMA), signed accumulation -----------------------
    wmma_pass(Gv + (size_t)(bN + j0) * GSTRIDE, &s_h[4096], s_y, lane, wave);
    __syncthreads();
    {
      float le = s_le[p];
      if (le != 0.f) {
        const float* y = &s_y[(jj*16 + ii) * YP];
        float sh0 = s_geo[p*4], sh1 = s_geo[p*4+1], sh2 = s_geo[p*4+2];
        float shv[3] = {sh0, sh1, sh2};
        float w = sqrtf(fabsf(le));
        float* acc = (le > 0.f) ? s_accP : s_accN;
        for (int o = 0; o < 16; ++o) {
          float vs = S_A * y[o] + S_B * (sh0*y[16+o] + sh1*y[32+o] + sh2*y[48+o]);
          atomicAdd(&acc[ii*28 + o], w * vs);
        }
        for (int o = 0; o < 4; ++o) {
          float ya = y[64 + o];
          for (int x = 0; x < 3; ++x) {
            float vv = S_A * shv[x] * ya + S_V2 * y[68 + 4*x + o];
            atomicAdd(&acc[ii*28 + 16 + o*3 + x], w * vv);
          }
        }
      }
    }
    __syncthreads();
  }
  // ---- write partials ------------------------------------------------------
  if (p < 16) zpart[(size_t)(bN + i0 + p) * JCHUNKS + jc] = s_accZ[p];
  for (int q = p; q < 16 * 28; q += 256) {
    int i = q / 28, o = q % 28;
    size_t ix = ((size_t)(bN + i0 + i) * JCHUNKS + jc) * 28 + o;
    outP[ix] = s_accP[q];
    outN[ix] = s_accN[q];
  }
}

// ---------------------------------------------------------------------------
// Kernel 4: reduce partials, apply sign-correct 1/sqrt(|z|), validity.
// ---------------------------------------------------------------------------
__global__ void k_final(const float* __restrict__ zpart,
                        const float* __restrict__ outP,
                        const float* __restrict__ outN,
                        const float* __restrict__ validv,
                        float* __restrict__ out) {
  int t = blockIdx.x * 256 + threadIdx.x;
  if (t >= BB * NN) return;
  float z = 0.f;
  for (int c = 0; c < JCHUNKS; ++c) z += zpart[(size_t)t * JCHUNKS + c];
  const float* src; float s;
  if (z == 0.f)      { src = outP; s = 1.f; }            // reference: z -> 1
  else if (z > 0.f)  { src = outP; s = 1.f / sqrtf(z); }
  else               { src = outN; s = 1.f / sqrtf(-z); }
  float v = validv[t];
  for (int o = 0; o < DIMF; ++o) {
    float a = 0.f;
    for (int c = 0; c < JCHUNKS; ++c) a += src[((size_t)t * JCHUNKS + c) * 28 + o];
    out[(size_t)t * DIMF + o] = v * s * a;
  }
}

// ---------------------------------------------------------------------------
extern "C" void kernel_launch(void* const* d_in, const int* in_sizes, int n_in,
                              void* d_out, int out_size, void* d_ws, size_t ws_size,
                              hipStream_t stream) {
  const float* f      = (const float*)d_in[0];
  const float* pos    = (const float*)d_in[1];
  const float* wq_s   = (const float*)d_in[2];
  const float* wq_v   = (const float*)d_in[3];
  const float* fck_w1 = (const float*)d_in[4];
  const float* fck_w2 = (const float*)d_in[5];
  const float* fcv_w1 = (const float*)d_in[6];
  const float* fcv_w2 = (const float*)d_in[7];
  const float* dot_ws = (const float*)d_in[8];
  const float* dot_wv = (const float*)d_in[9];

  // workspace layout (floats): total ~8.7 MB
  float* ws     = (float*)d_ws;
  float* Gk     = ws;                 // 983040
  float* Gv     = Gk + 983040;        // 983040
  float* qds    = Gv + 983040;        // 12288
  float* qdv    = qds + 12288;        // 9216
  float* validv = qdv + 9216;         // 768
  float* zpart  = validv + 768;       // 3072
  float* outP   = zpart + 3072;       // 86016
  float* outN   = outP + 86016;       // 86016
  float* out    = (float*)d_out;

  k_prep_g<<<(BB*NN*16*80 + 255)/256, 256, 0, stream>>>(f, fck_w2, fcv_w2, Gk, Gv);
  k_prep_q<<<(BB*NN + 255)/256, 256, 0, stream>>>(f, wq_s, wq_v, dot_ws, dot_wv,
                                                  qds, qdv, validv);
  k_attn<<<dim3(JCHUNKS, NN/16, BB), 256, 0, stream>>>(pos, fck_w1, fcv_w1,
                                                       Gk, Gv, qds, qdv, validv,
                                                       zpart, outP, outN);
  k_final<<<(BB*NN + 255)/256, 256, 0, stream>>>(zpart, outP, outN, validv, out);
}